// _SparseConv_78700980731975
// MI455X (gfx1250) — compile-verified
//
#include <hip/hip_runtime.h>

typedef float v2f __attribute__((ext_vector_type(2)));
typedef float v8f __attribute__((ext_vector_type(8)));

#define C_IN   16
#define C_OUT  32
#define NOFF   27               // 3*3*3 kernel taps
#define ODIM   126              // output spatial dim (128 - 3 + 1)
#define KELEMS (NOFF * C_IN * C_OUT)   // 13824 floats = 55296 B LDS

// ---------------------------------------------------------------------------
// Phase 0: zero the dense output (256 MB). Vectorized grid-stride.
// ---------------------------------------------------------------------------
__global__ void __launch_bounds__(256)
zero_kernel(float* __restrict__ out, int n) {
    int gid    = blockIdx.x * blockDim.x + threadIdx.x;
    int stride = gridDim.x * blockDim.x;
    int n4 = n >> 2;
    float4* o4 = reinterpret_cast<float4*>(out);
    float4 z = make_float4(0.f, 0.f, 0.f, 0.f);
    for (int i = gid; i < n4; i += stride) o4[i] = z;
    for (int i = (n4 << 2) + gid; i < n; i += stride) out[i] = 0.f;
}

// ---------------------------------------------------------------------------
// Phase 1: scatter sparse conv. One wave32 handles 16 points.
// Per kernel tap: D(16x32) = A(16x16) * B(16x32) via 8x v_wmma_f32_16x16x4_f32,
// then bounds-checked global_atomic_add_f32 scatter of the 16x32 tile.
// ---------------------------------------------------------------------------
__global__ void __launch_bounds__(128)
sparse_conv_kernel(const float* __restrict__ values,
                   const float* __restrict__ kernel,
                   const int*   __restrict__ indices,
                   float*       __restrict__ out,
                   int npts, int ngroups) {
    __shared__ float sk[KELEMS];
    // Stage 27x16x32 weights into LDS (flat copy; layout off*512 + cin*32 + cout)
    for (int i = threadIdx.x; i < KELEMS; i += 128) sk[i] = kernel[i];
    __syncthreads();

    const int wave  = threadIdx.x >> 5;
    const int group = blockIdx.x * 4 + wave;
    if (group >= ngroups) return;   // wave-uniform: EXEC stays all-ones below

    const int lane   = threadIdx.x & 31;
    const int laneLo = lane & 15;
    const int hi     = lane >> 4;   // 0: lanes 0-15, 1: lanes 16-31

    // ---- A fragments: 16 points x 16 c_in, f32 16x16x4 A layout.
    // VGPR pair per K-chunk c: lanes 0-15 hold K=4c+0,4c+1 ; lanes 16-31 hold K=4c+2,4c+3
    v2f a[4];
    {
        int p = group * 16 + laneLo;
        if (p < npts) {
            const float* vp = values + p * C_IN;
            #pragma unroll
            for (int c = 0; c < 4; ++c) {
                int cb = 4 * c + 2 * hi;
                a[c].x = vp[cb];
                a[c].y = vp[cb + 1];
            }
        } else {
            #pragma unroll
            for (int c = 0; c < 4; ++c) { a[c].x = 0.f; a[c].y = 0.f; }
        }
    }

    // ---- Coordinates of this lane's 8 D-rows (rows hi*8 .. hi*8+7).
    int pd[8], ph[8], pw[8];
    #pragma unroll
    for (int r = 0; r < 8; ++r) {
        int q = group * 16 + hi * 8 + r;
        if (q < npts) {
            pd[r] = indices[3 * q + 0];
            ph[r] = indices[3 * q + 1];
            pw[r] = indices[3 * q + 2];
        } else {
            pd[r] = 1 << 28; ph[r] = 1 << 28; pw[r] = 1 << 28;  // always OOB
        }
    }

    for (int off = 0; off < NOFF; ++off) {
        int kd = off / 9;
        int r9 = off - 9 * kd;
        int kh = r9 / 3;
        int kw = r9 - 3 * kh;

        const float* kb = sk + off * (C_IN * C_OUT);

        v8f acc0 = {};   // cout 0..15
        v8f acc1 = {};   // cout 16..31
        #pragma unroll
        for (int c = 0; c < 4; ++c) {
            int cb = 4 * c + 2 * hi;   // B row K for this lane half
            v2f b0, b1;
            b0.x = kb[cb       * C_OUT + laneLo];
            b0.y = kb[(cb + 1) * C_OUT + laneLo];
            b1.x = kb[cb       * C_OUT + 16 + laneLo];
            b1.y = kb[(cb + 1) * C_OUT + 16 + laneLo];
            acc0 = __builtin_amdgcn_wmma_f32_16x16x4_f32(
                false, a[c], false, b0, (short)0, acc0, false, false);
            acc1 = __builtin_amdgcn_wmma_f32_16x16x4_f32(
                false, a[c], false, b1, (short)0, acc1, false, false);
        }

        // ---- Scatter: D VGPR r -> row (hi*8 + r), col laneLo (+16 for tile 1)
        #pragma unroll
        for (int r = 0; r < 8; ++r) {
            int od = pd[r] - kd;
            int oh = ph[r] - kh;
            int ow = pw[r] - kw;
            if ((unsigned)od < ODIM && (unsigned)oh < ODIM && (unsigned)ow < ODIM) {
                int base = ((od * ODIM + oh) * ODIM + ow) * C_OUT + laneLo;
                unsafeAtomicAdd(out + base,      acc0[r]);
                unsafeAtomicAdd(out + base + 16, acc1[r]);
            }
        }
    }
}

// ---------------------------------------------------------------------------
extern "C" void kernel_launch(void* const* d_in, const int* in_sizes, int n_in,
                              void* d_out, int out_size, void* d_ws, size_t ws_size,
                              hipStream_t stream) {
    const float* values  = (const float*)d_in[0];
    const float* kernel  = (const float*)d_in[1];
    const int*   indices = (const int*)  d_in[2];
    float*       out     = (float*)d_out;

    int npts    = in_sizes[0] / C_IN;        // 20000
    int ngroups = (npts + 15) / 16;          // 1250 waves of work

    zero_kernel<<<2048, 256, 0, stream>>>(out, out_size);

    int blocks = (ngroups + 3) / 4;          // 4 waves (128 threads) per block
    sparse_conv_kernel<<<blocks, 128, 0, stream>>>(values, kernel, indices, out,
                                                   npts, ngroups);
}